// RACE2T_15229954031687
// MI455X (gfx1250) — compile-verified
//
#include <hip/hip_runtime.h>
#include <hip/hip_bf16.h>
#include <math.h>

typedef __attribute__((ext_vector_type(2))) float v2f;
typedef __attribute__((ext_vector_type(8))) float v8f;

#define N_ENT   50000
#define N_REL   237
#define N_TYPE  4000
#define DE      100
#define NOUT    128
#define DT      200
#define N_EDGE  800000
#define BATCH   4096
#define ALPHA   0.2f
#define EPS     1e-5f
#define NUM_FILT 32
#define FILT_W   9
#define CONV_OW  60
#define POOL_OW  30
#define FC_LEN   960

// ---- workspace layout (float offsets) ----
#define OFF_SMALL  0                       // 4096 floats of scalars/vectors
#define OFF_HSCORE 4096
#define OFF_EWA2   (OFF_HSCORE + N_ENT)
#define OFF_SMAX   (OFF_EWA2  + N_ENT)
#define OFF_DENOM  (OFF_SMAX  + N_ENT)
#define OFF_E      (OFF_DENOM + N_ENT)                 // 4096*128
#define OFF_XFC    (OFF_E     + BATCH*NOUT)            // 4096*200
#define OFF_POOL   (OFF_XFC   + BATCH*DT)              // 4096*960
#define OFF_UNION  (OFF_POOL  + BATCH*FC_LEN)
#define OFF_EX     OFF_UNION                           // 800000 (graph phase)
#define OFF_AGGC   (OFF_EX + N_EDGE)                   // 50000*100 (graph phase)
#define OFF_CONV   OFF_UNION                           // 4096*32*60 (cnn phase, aliases graph bufs)

// ---- SMALL sub-offsets ----
#define SM_WA1   0
#define SM_WA2   128
#define SM_RWA2  256
#define SM_BN1   512    // [0]=sum [1]=sumsq
#define SM_S1    514
#define SM_T1    515
#define SM_F3SUM 520    // 32
#define SM_F3SQ  552    // 32
#define SM_S3    584    // 32
#define SM_T3    616    // 32
#define SM_S2    648    // 200
#define SM_T2    848    // 200

__device__ __forceinline__ void atomicMaxF(float* addr, float val) {
    int* ia = (int*)addr;
    int old = __float_as_int(*addr);
    while (__int_as_float(old) < val) {
        int assumed = old;
        old = atomicCAS(ia, assumed, __float_as_int(val));
        if (old == assumed) break;
    }
}

__global__ void k_fill_neginf(float* __restrict__ p, int n) {
    int i = blockIdx.x * blockDim.x + threadIdx.x;
    if (i < n) p[i] = -INFINITY;
}

// wa1 = W_f @ a[:128], wa2 = W_f @ a[128:], rwa2[r] = R_w[r] . wa2
__global__ void k_prep(const float* __restrict__ W_f, const float* __restrict__ a_vec,
                       const float* __restrict__ R_w, float* __restrict__ small) {
    __shared__ float s_a[2 * NOUT];
    __shared__ float s_wa2[DE];
    int tid = threadIdx.x;
    if (tid < 2 * NOUT) s_a[tid] = a_vec[tid];
    __syncthreads();
    if (tid < DE) {
        float a1 = 0.f, a2 = 0.f;
        for (int j = 0; j < NOUT; ++j) {
            float w = W_f[tid * NOUT + j];
            a1 += w * s_a[j];
            a2 += w * s_a[NOUT + j];
        }
        small[SM_WA1 + tid] = a1;
        small[SM_WA2 + tid] = a2;
        s_wa2[tid] = a2;
    }
    __syncthreads();
    if (tid < N_REL) {
        float acc = 0.f;
        for (int k = 0; k < DE; ++k) acc += R_w[tid * DE + k] * s_wa2[k];
        small[SM_RWA2 + tid] = acc;
    }
}

// hscore[n] = E_w[n].wa1 ; ewa2[n] = E_w[n].wa2
__global__ void k_entity(const float* __restrict__ E_w, const float* __restrict__ small,
                         float* __restrict__ hscore, float* __restrict__ ewa2) {
    __shared__ float s_w[2 * DE];
    int tid = threadIdx.x;
    if (tid < DE)           s_w[tid]      = small[SM_WA1 + tid];
    else if (tid < 2 * DE)  s_w[tid]      = small[SM_WA2 + tid - DE];
    __syncthreads();
    int n = blockIdx.x * blockDim.x + tid;
    if (n < N_ENT) {
        const float* row = E_w + (size_t)n * DE;
        float a1 = 0.f, a2 = 0.f;
        for (int k = 0; k < DE; ++k) { float v = row[k]; a1 += v * s_w[k]; a2 += v * s_w[DE + k]; }
        hscore[n] = a1;
        ewa2[n]   = a2;
    }
}

__global__ void k_score(const int* __restrict__ ei, const int* __restrict__ et,
                        const float* __restrict__ hscore, const float* __restrict__ ewa2,
                        const float* __restrict__ small,
                        float* __restrict__ score, float* __restrict__ smax) {
    int e = blockIdx.x * blockDim.x + threadIdx.x;
    if (e >= N_EDGE) return;
    int src = ei[e], dst = ei[N_EDGE + e], r = et[e];
    float s = hscore[dst] + ewa2[src] - small[SM_RWA2 + r];
    s = s > 0.f ? s : ALPHA * s;       // leaky relu
    score[e] = s;
    atomicMaxF(&smax[dst], s);
}

__global__ void k_expsum(const int* __restrict__ ei, float* __restrict__ score,
                         const float* __restrict__ smax, float* __restrict__ denom) {
    int e = blockIdx.x * blockDim.x + threadIdx.x;
    if (e >= N_EDGE) return;
    int dst = ei[N_EDGE + e];
    float ex = expf(score[e] - smax[dst]);
    score[e] = ex;
    atomicAdd(&denom[dst], ex);
}

// one wave per edge: aggcomp[dst] += att * (E_w[src] - R_w[et])
__global__ void k_agg(const int* __restrict__ ei, const int* __restrict__ etyp,
                      const float* __restrict__ E_w, const float* __restrict__ R_w,
                      const float* __restrict__ ex, const float* __restrict__ denom,
                      float* __restrict__ aggc) {
    int gid = blockIdx.x * blockDim.x + threadIdx.x;
    int e = gid >> 5;
    int lane = gid & 31;
    if (e >= N_EDGE) return;
    int src = ei[e], dst = ei[N_EDGE + e], r = etyp[e];
    float att = ex[e] / (denom[dst] + 1e-16f);
    const float* er = E_w + (size_t)src * DE;
    const float* rr = R_w + (size_t)r * DE;
    float* ar = aggc + (size_t)dst * DE;
    for (int d = lane; d < DE; d += 32)
        atomicAdd(&ar[d], att * (er[d] - rr[d]));
}

// e[b,:] = elu( aggcomp[x_batch[b],:] @ W_f )   gathered WMMA GEMM [4096,100]x[100,128]
__global__ void k_egemm(const int* __restrict__ xb, const float* __restrict__ aggc,
                        const float* __restrict__ W_f, float* __restrict__ eout) {
    int wave = (blockIdx.x * blockDim.x + threadIdx.x) >> 5;
    int lane = threadIdx.x & 31;
    int tile_m = wave >> 3, tile_n = wave & 7;
    int m0 = tile_m * 16, n0 = tile_n * 16;
    int half = lane >> 4, l15 = lane & 15, koff = half * 2;
    int row = xb[m0 + l15];
    const float* ap = aggc + (size_t)row * DE;
    const float* bp = W_f + n0 + l15;
    v8f c = (v8f)(0.0f);
    for (int k0 = 0; k0 < DE; k0 += 4) {
        v2f a = { ap[k0 + koff], ap[k0 + koff + 1] };
        v2f b = { bp[(size_t)(k0 + koff) * NOUT], bp[(size_t)(k0 + koff + 1) * NOUT] };
        c = __builtin_amdgcn_wmma_f32_16x16x4_f32(false, a, false, b, (short)0, c, false, false);
    }
    int rbase = m0 + half * 8;
    int col = n0 + l15;
    for (int v = 0; v < 8; ++v) {
        float x = c[v];
        x = x > 0.f ? x : (expf(x) - 1.f);   // elu
        eout[(size_t)(rbase + v) * NOUT + col] = x;
    }
}

__global__ void k_bn1_stats(const float* __restrict__ x, int n, float* __restrict__ small) {
    __shared__ float ssum[256], ssq[256];
    int tid = threadIdx.x;
    int i = blockIdx.x * blockDim.x + tid;
    float v = (i < n) ? x[i] : 0.f;
    ssum[tid] = v; ssq[tid] = v * v;
    __syncthreads();
    for (int s = 128; s > 0; s >>= 1) {
        if (tid < s) { ssum[tid] += ssum[tid + s]; ssq[tid] += ssq[tid + s]; }
        __syncthreads();
    }
    if (tid == 0) { atomicAdd(&small[SM_BN1], ssum[0]); atomicAdd(&small[SM_BN1 + 1], ssq[0]); }
}

__global__ void k_bn1_final(float* __restrict__ small, const float* __restrict__ g,
                            const float* __restrict__ b) {
    float n = (float)(BATCH * NOUT);
    float mean = small[SM_BN1] / n;
    float var  = small[SM_BN1 + 1] / n - mean * mean;
    float s = g[0] * rsqrtf(var + EPS);
    small[SM_S1] = s;
    small[SM_T1] = b[0] - mean * s;
}

// conv(stride 2, width 9) with bn1 fused on input; also accumulates bn3 stats
__global__ void k_conv(const float* __restrict__ e, float* __restrict__ small,
                       const float* __restrict__ cw, const float* __restrict__ cb,
                       float* __restrict__ convout) {
    __shared__ float sSum[NUM_FILT], sSq[NUM_FILT];
    int tid = threadIdx.x;
    if (tid < NUM_FILT) { sSum[tid] = 0.f; sSq[tid] = 0.f; }
    __syncthreads();
    int i = blockIdx.x * blockDim.x + tid;
    if (i < BATCH * NUM_FILT * CONV_OW) {
        int b = i / (NUM_FILT * CONV_OW);
        int rem = i - b * (NUM_FILT * CONV_OW);
        int f = rem / CONV_OW, w = rem - f * CONV_OW;
        float s1 = small[SM_S1], t1 = small[SM_T1];
        const float* er = e + (size_t)b * NOUT + 2 * w;
        float acc = cb[f];
        for (int k = 0; k < FILT_W; ++k) acc += cw[f * FILT_W + k] * (s1 * er[k] + t1);
        convout[i] = acc;
        atomicAdd(&sSum[f], acc);
        atomicAdd(&sSq[f], acc * acc);
    }
    __syncthreads();
    if (tid < NUM_FILT) {
        atomicAdd(&small[SM_F3SUM + tid], sSum[tid]);
        atomicAdd(&small[SM_F3SQ + tid],  sSq[tid]);
    }
}

__global__ void k_bn3_final(float* __restrict__ small, const float* __restrict__ g,
                            const float* __restrict__ b) {
    int f = threadIdx.x;
    if (f < NUM_FILT) {
        float n = (float)(BATCH * CONV_OW);
        float mean = small[SM_F3SUM + f] / n;
        float var  = small[SM_F3SQ + f] / n - mean * mean;
        float s = g[f] * rsqrtf(var + EPS);
        small[SM_S3 + f] = s;
        small[SM_T3 + f] = b[f] - mean * s;
    }
}

__global__ void k_pool(const float* __restrict__ conv, const float* __restrict__ small,
                       float* __restrict__ pooled) {
    int i = blockIdx.x * blockDim.x + threadIdx.x;
    if (i >= BATCH * NUM_FILT * POOL_OW) return;
    int b = i / (NUM_FILT * POOL_OW);
    int rem = i - b * (NUM_FILT * POOL_OW);
    int f = rem / POOL_OW, wp = rem - f * POOL_OW;
    const float* cr = conv + (size_t)b * (NUM_FILT * CONV_OW) + f * CONV_OW + 2 * wp;
    float s = small[SM_S3 + f], t = small[SM_T3 + f];
    float a0 = fmaxf(s * cr[0] + t, 0.f);
    float a1 = fmaxf(s * cr[1] + t, 0.f);
    pooled[(size_t)b * FC_LEN + f * POOL_OW + wp] = fmaxf(a0, a1);
}

// xfc = pooled @ fc_w^T + fc_b   WMMA [4096,960]x[960,200] (N padded to 208)
#define FC_NTILES 13
__global__ void k_fc(const float* __restrict__ pooled, const float* __restrict__ fc_w,
                     const float* __restrict__ fc_b, float* __restrict__ xfc) {
    int wave = (blockIdx.x * blockDim.x + threadIdx.x) >> 5;
    int lane = threadIdx.x & 31;
    int tile_m = wave / FC_NTILES;
    int tile_n = wave - tile_m * FC_NTILES;
    int m0 = tile_m * 16, n0 = tile_n * 16;
    int half = lane >> 4, l15 = lane & 15, koff = half * 2;
    const float* ap = pooled + (size_t)(m0 + l15) * FC_LEN;
    int ncol = n0 + l15;
    bool nok = ncol < DT;
    const float* bp = fc_w + (size_t)(nok ? ncol : 0) * FC_LEN;
    v8f c = (v8f)(0.0f);
    for (int k0 = 0; k0 < FC_LEN; k0 += 4) {
        v2f a = { ap[k0 + koff], ap[k0 + koff + 1] };
        v2f b;
        b.x = nok ? bp[k0 + koff] : 0.f;
        b.y = nok ? bp[k0 + koff + 1] : 0.f;
        c = __builtin_amdgcn_wmma_f32_16x16x4_f32(false, a, false, b, (short)0, c, false, false);
    }
    if (nok) {
        float bias = fc_b[ncol];
        int rbase = m0 + half * 8;
        for (int v = 0; v < 8; ++v)
            xfc[(size_t)(rbase + v) * DT + ncol] = c[v] + bias;
    }
}

__global__ void k_bn2_stats(const float* __restrict__ xfc, float* __restrict__ small,
                            const float* __restrict__ g, const float* __restrict__ b) {
    __shared__ float ssum[256], ssq[256];
    int t = blockIdx.x;
    int tid = threadIdx.x;
    float a = 0.f, q = 0.f;
    for (int r = tid; r < BATCH; r += 256) {
        float v = xfc[(size_t)r * DT + t];
        a += v; q += v * v;
    }
    ssum[tid] = a; ssq[tid] = q;
    __syncthreads();
    for (int s = 128; s > 0; s >>= 1) {
        if (tid < s) { ssum[tid] += ssum[tid + s]; ssq[tid] += ssq[tid + s]; }
        __syncthreads();
    }
    if (tid == 0) {
        float n = (float)BATCH;
        float mean = ssum[0] / n, var = ssq[0] / n - mean * mean;
        float s2 = g[t] * rsqrtf(var + EPS);
        small[SM_S2 + t] = s2;
        small[SM_T2 + t] = b[t] - mean * s2;
    }
}

// out = sigmoid( relu(bn2(xfc)) @ T_w^T + b_bias )  WMMA [4096,200]x[200,4000]
#define OUT_NTILES 250
__global__ void k_logits(const float* __restrict__ xfc, const float* __restrict__ small,
                         const float* __restrict__ T_w, const float* __restrict__ bb,
                         float* __restrict__ out) {
    int wave = (blockIdx.x * blockDim.x + threadIdx.x) >> 5;
    int lane = threadIdx.x & 31;
    int tile_m = wave / OUT_NTILES;
    int tile_n = wave - tile_m * OUT_NTILES;
    int m0 = tile_m * 16, n0 = tile_n * 16;
    int half = lane >> 4, l15 = lane & 15, koff = half * 2;
    const float* ap = xfc + (size_t)(m0 + l15) * DT;
    int ncol = n0 + l15;
    const float* bp = T_w + (size_t)ncol * DT;
    v8f c = (v8f)(0.0f);
    for (int k0 = 0; k0 < DT; k0 += 4) {
        int ka = k0 + koff;
        v2f a;
        a.x = fmaxf(small[SM_S2 + ka]     * ap[ka]     + small[SM_T2 + ka],     0.f);
        a.y = fmaxf(small[SM_S2 + ka + 1] * ap[ka + 1] + small[SM_T2 + ka + 1], 0.f);
        v2f b = { bp[ka], bp[ka + 1] };
        c = __builtin_amdgcn_wmma_f32_16x16x4_f32(false, a, false, b, (short)0, c, false, false);
    }
    float bias = bb[ncol];
    int rbase = m0 + half * 8;
    for (int v = 0; v < 8; ++v) {
        float l = c[v] + bias;
        float y = 1.f / (1.f + expf(-l));
        __builtin_nontemporal_store(y, &out[(size_t)(rbase + v) * N_TYPE + ncol]);
    }
}

extern "C" void kernel_launch(void* const* d_in, const int* in_sizes, int n_in,
                              void* d_out, int out_size, void* d_ws, size_t ws_size,
                              hipStream_t stream) {
    (void)in_sizes; (void)n_in; (void)out_size; (void)ws_size;
    const int*   x_batch    = (const int*)d_in[0];
    const int*   edge_index = (const int*)d_in[1];
    const int*   edge_type  = (const int*)d_in[2];
    const float* E_w   = (const float*)d_in[3];
    const float* R_w   = (const float*)d_in[4];
    const float* T_w   = (const float*)d_in[5];
    const float* W_f   = (const float*)d_in[6];
    const float* a_vec = (const float*)d_in[7];
    const float* cw    = (const float*)d_in[8];
    const float* cb    = (const float*)d_in[9];
    const float* bn1_g = (const float*)d_in[10];
    const float* bn1_b = (const float*)d_in[11];
    const float* bn3_g = (const float*)d_in[12];
    const float* bn3_b = (const float*)d_in[13];
    const float* bn2_g = (const float*)d_in[14];
    const float* bn2_b = (const float*)d_in[15];
    const float* fc_w  = (const float*)d_in[16];
    const float* fc_b  = (const float*)d_in[17];
    const float* bbias = (const float*)d_in[18];
    float* out = (float*)d_out;
    float* ws  = (float*)d_ws;

    float* small  = ws + OFF_SMALL;
    float* hscore = ws + OFF_HSCORE;
    float* ewa2   = ws + OFF_EWA2;
    float* smax   = ws + OFF_SMAX;
    float* denom  = ws + OFF_DENOM;
    float* ebuf   = ws + OFF_E;
    float* xfc    = ws + OFF_XFC;
    float* pooled = ws + OFF_POOL;
    float* exsc   = ws + OFF_EX;
    float* aggc   = ws + OFF_AGGC;
    float* convb  = ws + OFF_CONV;

    // per-call (re)initialization — deterministic, graph-capture safe
    (void)hipMemsetAsync(small, 0, 4096 * sizeof(float), stream);
    (void)hipMemsetAsync(denom, 0, (size_t)N_ENT * sizeof(float), stream);
    (void)hipMemsetAsync(aggc,  0, (size_t)N_ENT * DE * sizeof(float), stream);
    k_fill_neginf<<<(N_ENT + 255) / 256, 256, 0, stream>>>(smax, N_ENT);

    // graph attention (refactored: W_f pulled out of the segment sum)
    k_prep  <<<1, 256, 0, stream>>>(W_f, a_vec, R_w, small);
    k_entity<<<(N_ENT + 255) / 256, 256, 0, stream>>>(E_w, small, hscore, ewa2);
    k_score <<<N_EDGE / 256, 256, 0, stream>>>(edge_index, edge_type, hscore, ewa2, small, exsc, smax);
    k_expsum<<<N_EDGE / 256, 256, 0, stream>>>(edge_index, exsc, smax, denom);
    k_agg   <<<N_EDGE / 8,   256, 0, stream>>>(edge_index, edge_type, E_w, R_w, exsc, denom, aggc);

    // gathered entity GEMM + elu  (WMMA f32)
    k_egemm<<<(BATCH / 16) * (NOUT / 16) / 8, 256, 0, stream>>>(x_batch, aggc, W_f, ebuf);

    // bn1 -> conv(+bn3 stats) -> bn3+relu+pool
    k_bn1_stats<<<(BATCH * NOUT) / 256, 256, 0, stream>>>(ebuf, BATCH * NOUT, small);
    k_bn1_final<<<1, 1, 0, stream>>>(small, bn1_g, bn1_b);
    k_conv<<<(BATCH * NUM_FILT * CONV_OW) / 256, 256, 0, stream>>>(ebuf, small, cw, cb, convb);
    k_bn3_final<<<1, 32, 0, stream>>>(small, bn3_g, bn3_b);
    k_pool<<<(BATCH * NUM_FILT * POOL_OW) / 256, 256, 0, stream>>>(convb, small, pooled);

    // fc GEMM (WMMA f32, N padded 200->208)
    k_fc<<<((BATCH / 16) * FC_NTILES) / 8, 256, 0, stream>>>(pooled, fc_w, fc_b, xfc);

    // bn2 column stats, then logits GEMM with fused bn2+relu on A, bias+sigmoid+NT-store epilogue
    k_bn2_stats<<<DT, 256, 0, stream>>>(xfc, small, bn2_g, bn2_b);
    k_logits<<<((BATCH / 16) * OUT_NTILES) / 8, 256, 0, stream>>>(xfc, small, T_w, bbias, out);
}